// PeakAreaFloodLoss_16612933501051
// MI455X (gfx1250) — compile-verified
//
#include <hip/hip_runtime.h>
#include <stdint.h>

#define ROW_S 8192
#define BLK   256
#define NEGV  (-1.0e30f)

template<int N>
__device__ __forceinline__ void wait_async() {
    asm volatile("s_wait_asynccnt %0" :: "i"(N) : "memory");
}

__device__ __forceinline__ void async_b128(uint32_t lds_off, const void* gaddr) {
    asm volatile("global_load_async_to_lds_b128 %0, %1, off"
                 :: "v"(lds_off), "v"(gaddr) : "memory");
}

__global__ __launch_bounds__(BLK) void flood_loss_kernel(
    const float* __restrict__ pred,
    const float* __restrict__ tgt,
    const int*   __restrict__ mask,
    float*       __restrict__ row_out,   // per-row loss (two-stage path), or null
    float*       __restrict__ out,       // scalar accumulator (atomic fallback)
    float inv_rows)
{
    extern __shared__ uint8_t smem[];
    float* s_p = (float*)(smem);            // 32 KB
    float* s_t = (float*)(smem + 32768);    // 32 KB
    int*   s_m = (int*)(smem + 65536);      // 32 KB (scratch after pass 1)

    const int tid = threadIdx.x;
    const size_t base = (size_t)blockIdx.x * ROW_S;
    const float* gp = pred + base;
    const float* gt = tgt + base;
    const int*   gm = mask + base;

    // LDS byte offsets (generic->local lowering truncates flat ptr to LDS offset)
    const uint32_t lp = (uint32_t)(uintptr_t)s_p;
    const uint32_t lt = (uint32_t)(uintptr_t)s_t;
    const uint32_t lm = (uint32_t)(uintptr_t)s_m;

    // ---- Issue async stage of the whole row: order (p_k, t_k, m_k), k=0..7 ----
    #pragma unroll
    for (int k = 0; k < 8; ++k) {
        const int c = tid + k * BLK;              // 16B chunk id
        const uint32_t off16 = (uint32_t)c * 16u;
        async_b128(lp + off16, gp + 4 * c);
        async_b128(lt + off16, gt + 4 * c);
        async_b128(lm + off16, gm + 4 * c);
    }

    // ---- Pass 1, pipelined: each thread consumes exactly the chunks it staged ----
    float sum = 0.0f, tmax = NEGV, pmax = NEGV;
    int n = 0;
    #pragma unroll
    for (int k = 0; k < 8; ++k) {
        switch (k) {   // chunk k ready when first 3*(k+1) async loads completed
            case 0: wait_async<21>(); break;
            case 1: wait_async<18>(); break;
            case 2: wait_async<15>(); break;
            case 3: wait_async<12>(); break;
            case 4: wait_async<9>();  break;
            case 5: wait_async<6>();  break;
            case 6: wait_async<3>();  break;
            default: wait_async<0>(); break;
        }
        const int c = tid + k * BLK;
        const float4 p4 = ((const float4*)s_p)[c];
        float4       t4 = ((const float4*)s_t)[c];
        const int4   m4 = ((const int4*)s_m)[c];

        const float pe[4] = {p4.x, p4.y, p4.z, p4.w};
        float       te[4] = {t4.x, t4.y, t4.z, t4.w};
        const int   me[4] = {m4.x, m4.y, m4.z, m4.w};
        #pragma unroll
        for (int j = 0; j < 4; ++j) {
            const float p = pe[j], t = te[j];
            const bool valid = (me[j] != 0) && (p == p) && (t == t);
            if (valid) {
                const float d = p - t;
                sum += d * d;
                n   += 1;
                tmax = fmaxf(tmax, t);
                pmax = fmaxf(pmax, p);
            }
            te[j] = valid ? t : NEGV;           // sentinel for pass 2
        }
        t4.x = te[0]; t4.y = te[1]; t4.z = te[2]; t4.w = te[3];
        ((float4*)s_t)[c] = t4;                 // own chunk only
    }

    // wave32 reductions
    #pragma unroll
    for (int o = 16; o > 0; o >>= 1) {
        sum  += __shfl_xor(sum, o);
        n    += __shfl_xor(n, o);
        tmax  = fmaxf(tmax, __shfl_xor(tmax, o));
        pmax  = fmaxf(pmax, __shfl_xor(pmax, o));
    }

    __syncthreads();                  // all pass-1 reads of s_m done; reuse as scratch
    float* rf = (float*)s_m;
    int*   ri = (int*)s_m;
    const int wid = tid >> 5, lid = tid & 31;
    if (lid == 0) {
        rf[wid]      = sum;
        rf[8 + wid]  = tmax;
        rf[16 + wid] = pmax;
        ri[24 + wid] = n;
    }
    __syncthreads();
    if (tid == 0) {
        float s = 0.0f, tm = NEGV, pm = NEGV; int nn = 0;
        #pragma unroll
        for (int w = 0; w < 8; ++w) {
            s  += rf[w];
            tm  = fmaxf(tm, rf[8 + w]);
            pm  = fmaxf(pm, rf[16 + w]);
            nn += ri[24 + w];
        }
        rf[32] = s; rf[33] = tm; rf[34] = pm; ri[35] = nn;
        rf[36] = (tm > 0.0f) ? 0.8f * tm : -1.0e29f;  // effective threshold
    }
    __syncthreads();
    const float thr = rf[36];

    // ---- Pass 2 (LDS-resident, own chunks): high-flow RMSE terms ----
    float hsum = 0.0f; int hn = 0;
    #pragma unroll
    for (int k = 0; k < 8; ++k) {
        const int c = tid + k * BLK;
        const float4 t4 = ((const float4*)s_t)[c];   // NEGV entries fail threshold
        const float4 p4 = ((const float4*)s_p)[c];
        const float te[4] = {t4.x, t4.y, t4.z, t4.w};
        const float pe[4] = {p4.x, p4.y, p4.z, p4.w};
        #pragma unroll
        for (int j = 0; j < 4; ++j) {
            if (te[j] >= thr) {
                const float d = pe[j] - te[j];
                hsum += d * d;
                hn   += 1;
            }
        }
    }
    #pragma unroll
    for (int o = 16; o > 0; o >>= 1) {
        hsum += __shfl_xor(hsum, o);
        hn   += __shfl_xor(hn, o);
    }
    if (lid == 0) { rf[40 + wid] = hsum; ri[48 + wid] = hn; }
    __syncthreads();

    if (tid == 0) {
        float hs = 0.0f; int hh = 0;
        #pragma unroll
        for (int w = 0; w < 8; ++w) { hs += rf[40 + w]; hh += ri[48 + w]; }
        const float s  = rf[32];
        const float tm = rf[33];
        const float pm = rf[34];
        const int   nn = ri[35];
        const float overall = (nn > 0) ? sqrtf(s / (float)nn)  : 0.0f;
        const float peak    = (nn > 0) ? fabsf(pm - tm)        : 0.0f;
        const float hrmse   = (hh > 0) ? sqrtf(hs / (float)hh) : 0.0f;
        const float loss = 0.5f * overall + 2.0f * peak + 1.0f * hrmse;
        if (row_out) row_out[blockIdx.x] = loss;          // deterministic path
        else         atomicAdd(out, loss * inv_rows);     // fallback
    }
}

__global__ __launch_bounds__(BLK) void finalize_kernel(
    const float* __restrict__ row_loss, float* __restrict__ out,
    int rows, float inv_rows)
{
    const int tid = threadIdx.x;
    float s = 0.0f;
    for (int i = tid; i < rows; i += BLK) s += row_loss[i];
    #pragma unroll
    for (int o = 16; o > 0; o >>= 1) s += __shfl_xor(s, o);
    __shared__ float parts[8];
    if ((tid & 31) == 0) parts[tid >> 5] = s;
    __syncthreads();
    if (tid == 0) {
        float tot = 0.0f;
        #pragma unroll
        for (int w = 0; w < 8; ++w) tot += parts[w];
        out[0] = tot * inv_rows;
    }
}

extern "C" void kernel_launch(void* const* d_in, const int* in_sizes, int n_in,
                              void* d_out, int out_size, void* d_ws, size_t ws_size,
                              hipStream_t stream) {
    (void)n_in; (void)out_size;
    const float* pred = (const float*)d_in[0];
    const float* tgt  = (const float*)d_in[1];
    const int*   mask = (const int*)d_in[2];
    float* out = (float*)d_out;

    const long long total = in_sizes[0];
    const int rows = (int)(total / ROW_S);   // 2048 for the reference shapes
    const float inv_rows = 1.0f / (float)rows;
    const size_t shmem = 3ull * ROW_S * sizeof(float);   // 96 KB dynamic LDS

    if (ws_size >= (size_t)rows * sizeof(float)) {
        float* row_loss = (float*)d_ws;
        flood_loss_kernel<<<rows, BLK, shmem, stream>>>(pred, tgt, mask,
                                                        row_loss, out, inv_rows);
        finalize_kernel<<<1, BLK, 0, stream>>>(row_loss, out, rows, inv_rows);
    } else {
        hipMemsetAsync(out, 0, sizeof(float), stream);
        flood_loss_kernel<<<rows, BLK, shmem, stream>>>(pred, tgt, mask,
                                                        nullptr, out, inv_rows);
    }
}